// EMRouting_32229434589758
// MI455X (gfx1250) — compile-verified
//
#include <hip/hip_runtime.h>

#define B_      16
#define NI      2048
#define NO      64
#define P_      16
#define CHUNK   64
#define NCHUNK  (NI / CHUNK)      // 32
#define EPSF    1e-7f
#define MIN_VAR 0.0005f

typedef __attribute__((ext_vector_type(2))) float v2f;
typedef __attribute__((ext_vector_type(4))) float v4f;
typedef __attribute__((ext_vector_type(8))) float v8f;

// ---------------------------------------------------------------------------
// Fused kernel: (optionally) recompute R from previous-iteration params, then
// accumulate per-chunk partial moments S0/S1/S2 with V_WMMA_F32_16X16X4_F32.
// Grid: (NCHUNK, B).  Block: 256 threads (8 wave32).
// votes layout: [B][Ni][No][P], contiguous in P.
// ---------------------------------------------------------------------------
template <bool FIRST>
__global__ __launch_bounds__(256) void em_fused(
    const float* __restrict__ votes, const float* __restrict__ acts,
    const float* __restrict__ mu_g, const float* __restrict__ iv_g,
    const float* __restrict__ base_g,
    float* __restrict__ pS0, float* __restrict__ pS1, float* __restrict__ pS2)
{
  __shared__ float s_mu[P_ * NO];           // transposed [p][o] (conflict-free)
  __shared__ float s_iv[P_ * NO];
  __shared__ float s_base[NO];
  __shared__ float s_rw[CHUNK * (NO + 1)];  // Rw[i][o], padded stride 65

  const int t     = threadIdx.x;
  const int b     = blockIdx.y;
  const int chunk = blockIdx.x;
  const int i0    = chunk * CHUNK;
  const float* vb = votes + ((size_t)b * NI + i0) * (NO * P_);
  const float* ab = acts + b * NI + i0;

  if (FIRST) {
    // Iteration 0: R is uniform 1/No, so Rw[i][o] = a_i / No.
    for (int k = t; k < CHUNK * NO; k += 256) {
      int il = k >> 6, o = k & 63;
      s_rw[il * (NO + 1) + o] = ab[il] * (1.0f / NO);
    }
    __syncthreads();
  } else {
    for (int k = t; k < NO * P_; k += 256) {
      int o = k >> 4, p = k & 15;
      s_mu[p * NO + o] = mu_g[(b * NO + o) * P_ + p];
      s_iv[p * NO + o] = iv_g[(b * NO + o) * P_ + p];
    }
    if (t < NO) s_base[t] = base_g[b * NO + t];
    __syncthreads();

    // Stage 1: scores s[i][o] = log(a_j+eps) - 0.5*sum_p log(var)
    //                           - 0.5*sum_p (v-mu)^2 * invvar
    for (int pass = 0; pass < (CHUNK * NO) / 256; ++pass) {
      int idx = pass * 256 + t;
      int il = idx >> 6, o = idx & 63;
      const float* vp = vb + ((size_t)il * NO + o) * P_;
      __builtin_prefetch(vp + 4 * NO * P_, 0, 1);   // -> global_prefetch_b8
      float acc = 0.0f;
      #pragma unroll
      for (int q = 0; q < 4; ++q) {
        v4f v = *(const v4f*)(vp + 4 * q);
        float d;
        d = v.x - s_mu[(4*q+0)*NO + o]; acc = fmaf(d*d, s_iv[(4*q+0)*NO + o], acc);
        d = v.y - s_mu[(4*q+1)*NO + o]; acc = fmaf(d*d, s_iv[(4*q+1)*NO + o], acc);
        d = v.z - s_mu[(4*q+2)*NO + o]; acc = fmaf(d*d, s_iv[(4*q+2)*NO + o], acc);
        d = v.w - s_mu[(4*q+3)*NO + o]; acc = fmaf(d*d, s_iv[(4*q+3)*NO + o], acc);
      }
      s_rw[il * (NO + 1) + o] = s_base[o] - 0.5f * acc;
    }
    __syncthreads();

    // Stage 2: softmax over o (per i), then scale by a_i -> Rw.
    // 4 lanes per row i, 16 strided o's per lane, shfl_xor reduction.
    {
      int il = t >> 2;
      int sub = t & 3;
      float vals[16];
      float mx = -3.4e38f;
      #pragma unroll
      for (int k = 0; k < 16; ++k) {
        float s = s_rw[il * (NO + 1) + sub + 4 * k];
        vals[k] = s;
        mx = fmaxf(mx, s);
      }
      mx = fmaxf(mx, __shfl_xor(mx, 1, 32));
      mx = fmaxf(mx, __shfl_xor(mx, 2, 32));
      float sum = 0.0f;
      #pragma unroll
      for (int k = 0; k < 16; ++k) { vals[k] = __expf(vals[k] - mx); sum += vals[k]; }
      sum += __shfl_xor(sum, 1, 32);
      sum += __shfl_xor(sum, 2, 32);
      float scale = ab[il] / sum;
      #pragma unroll
      for (int k = 0; k < 16; ++k)
        s_rw[il * (NO + 1) + sub + 4 * k] = vals[k] * scale;
    }
    __syncthreads();
  }

  // Stage 3: per-chunk moments via WMMA.  Wave w owns o = 8w..8w+7.
  // A[m=p, k] = votes[ibase+k][o][p]   (lane = M, K = r + 2*(lane/16))
  // B[k, n]   = Rw[ibase+k][o]         (broadcast over N)
  // => every column of C accumulates S[o][p]; col 0 lives in lanes 0 & 16.
  const int lane = t & 31;
  const int wv = t >> 5;
  const int lh = lane >> 4;
  const int lm = lane & 15;
  const size_t pbase = ((size_t)chunk * B_ + b) * NO;

  for (int oi = 0; oi < 8; ++oi) {
    const int o = wv * 8 + oi;

    // S0 partial: wave-reduce Rw over the 64 i's of this chunk.
    float r0 = s_rw[lane * (NO + 1) + o] + s_rw[(lane + 32) * (NO + 1) + o];
    r0 += __shfl_xor(r0, 16, 32);
    r0 += __shfl_xor(r0, 8, 32);
    r0 += __shfl_xor(r0, 4, 32);
    r0 += __shfl_xor(r0, 2, 32);
    r0 += __shfl_xor(r0, 1, 32);

    v8f c1 = {0.f,0.f,0.f,0.f,0.f,0.f,0.f,0.f};
    v8f c2 = {0.f,0.f,0.f,0.f,0.f,0.f,0.f,0.f};
    #pragma unroll
    for (int s = 0; s < CHUNK / 4; ++s) {
      const int ibase = 4 * s + 2 * lh;
      const float* va = vb + ((size_t)ibase * NO + o) * P_ + lm;
      v2f A, A2, Bv;
      A.x  = va[0];
      A.y  = va[NO * P_];
      Bv.x = s_rw[ibase * (NO + 1) + o];
      Bv.y = s_rw[(ibase + 1) * (NO + 1) + o];
      A2.x = A.x * A.x;
      A2.y = A.y * A.y;
      c1 = __builtin_amdgcn_wmma_f32_16x16x4_f32(false, A,  false, Bv, (short)0, c1, false, false);
      c2 = __builtin_amdgcn_wmma_f32_16x16x4_f32(false, A2, false, Bv, (short)0, c2, false, false);
    }
    if (lm == 0) {  // column N=0: lane 0 -> p=0..7, lane 16 -> p=8..15
      float* o1 = pS1 + (pbase + o) * P_ + lh * 8;
      float* o2 = pS2 + (pbase + o) * P_ + lh * 8;
      #pragma unroll
      for (int r = 0; r < 8; ++r) { o1[r] = c1[r]; o2[r] = c2[r]; }
      if (lane == 0) pS0[pbase + o] = r0;
    }
  }
}

// ---------------------------------------------------------------------------
// Finalize: deterministically reduce chunk partials, compute mu/var/a_j and
// the params (mu, 1/var, softmax bias) for the next iteration.
// 16384 threads: one per (b,o,p).
// ---------------------------------------------------------------------------
__global__ __launch_bounds__(256) void em_finalize(
    const float* __restrict__ pS0, const float* __restrict__ pS1,
    const float* __restrict__ pS2,
    float* __restrict__ mu_g, float* __restrict__ iv_g,
    float* __restrict__ base_g,
    const float* __restrict__ act_bias_p, const float* __restrict__ sigma_bias_p,
    float beta, int final, float* __restrict__ out_mu, float* __restrict__ out_aj)
{
  const int g = blockIdx.x * 256 + threadIdx.x;
  const int cell = g >> 4;   // b*NO + o
  const int p = g & 15;

  float s0 = 0.f, s1 = 0.f, s2 = 0.f;
  for (int c = 0; c < NCHUNK; ++c) {
    const size_t pb = (size_t)c * (B_ * NO) + cell;
    s0 += pS0[pb];
    s1 += pS1[pb * P_ + p];
    s2 += pS2[pb * P_ + p];
  }
  const float mass = s0 + EPSF;
  const float mu = s1 / mass;
  // sum Rw*(v-mu)^2 = S2 - 2*mu*S1 + mu^2*S0
  const float var = (s2 - 2.0f * mu * s1 + mu * mu * s0) / mass + MIN_VAR;
  const float logv = __logf(var);
  float csum = (*sigma_bias_p + 0.5f * logv) * mass;
  float lsum = logv;
  #pragma unroll
  for (int d = 1; d < 16; d <<= 1) {
    csum += __shfl_xor(csum, d, 32);
    lsum += __shfl_xor(lsum, d, 32);
  }
  const float aj = 1.0f / (1.0f + __expf(-beta * (*act_bias_p - csum)));

  mu_g[cell * P_ + p] = mu;
  iv_g[cell * P_ + p] = 1.0f / var;
  if (p == 0) base_g[cell] = __logf(aj + EPSF) - 0.5f * lsum;
  if (final) {
    out_mu[cell * P_ + p] = mu;
    if (p == 0) out_aj[cell] = aj;
  }
}

// ---------------------------------------------------------------------------
extern "C" void kernel_launch(void* const* d_in, const int* in_sizes, int n_in,
                              void* d_out, int out_size, void* d_ws, size_t ws_size,
                              hipStream_t stream) {
  const float* votes      = (const float*)d_in[0];
  const float* acts       = (const float*)d_in[1];
  const float* act_bias   = (const float*)d_in[2];  // device scalar
  const float* sigma_bias = (const float*)d_in[3];  // device scalar

  float* ws = (float*)d_ws;
  const size_t n0 = (size_t)NCHUNK * B_ * NO;       // 32768
  const size_t n1 = n0 * P_;                        // 524288
  float* pS0    = ws;
  float* pS1    = pS0 + n0;
  float* pS2    = pS1 + n1;
  float* mu_g   = pS2 + n1;
  float* iv_g   = mu_g + (size_t)B_ * NO * P_;
  float* base_g = iv_g + (size_t)B_ * NO * P_;

  float* out    = (float*)d_out;
  float* out_mu = out;
  float* out_aj = out + B_ * NO * P_;

  const dim3 grid(NCHUNK, B_);
  const int fin_blocks = (B_ * NO * P_) / 256;      // 64
  // beta = 1 - 0.95^(it+1)
  const float b0 = 0.05f, b1 = 0.0975f, b2 = 0.142625f;

  // iter 0: uniform R stats -> params
  em_fused<true><<<grid, 256, 0, stream>>>(votes, acts, mu_g, iv_g, base_g,
                                           pS0, pS1, pS2);
  em_finalize<<<fin_blocks, 256, 0, stream>>>(pS0, pS1, pS2, mu_g, iv_g, base_g,
                                              act_bias, sigma_bias, b0, 0,
                                              out_mu, out_aj);
  // iter 1: fused R-update + stats
  em_fused<false><<<grid, 256, 0, stream>>>(votes, acts, mu_g, iv_g, base_g,
                                            pS0, pS1, pS2);
  em_finalize<<<fin_blocks, 256, 0, stream>>>(pS0, pS1, pS2, mu_g, iv_g, base_g,
                                              act_bias, sigma_bias, b1, 0,
                                              out_mu, out_aj);
  // iter 2: fused R-update + stats, final outputs
  em_fused<false><<<grid, 256, 0, stream>>>(votes, acts, mu_g, iv_g, base_g,
                                            pS0, pS1, pS2);
  em_finalize<<<fin_blocks, 256, 0, stream>>>(pS0, pS1, pS2, mu_g, iv_g, base_g,
                                              act_bias, sigma_bias, b2, 1,
                                              out_mu, out_aj);
}